// EntmaxNsect_79654463472378
// MI455X (gfx1250) — compile-verified
//
#include <hip/hip_runtime.h>
#include <cstdint>

// ---------------- problem constants (match reference) ----------------
#define ROWS        8192
#define DCOLS       8192
#define THREADS     256           // 8 waves of 32
#define CHUNKS      8             // float4 chunks per thread
#define PAIRS       16            // float2 pairs per thread (32 floats)
#define N_ITER      5

// alpha = 1.5 -> (alpha-1) = 0.5, exponent = 2
#define ALPHA_M1    0.5f
// (1/d)^(alpha-1) = (1/8192)^0.5
#define HI_OFF      0.011048543456039806f

typedef __attribute__((ext_vector_type(2))) float f2;
typedef __attribute__((ext_vector_type(4))) float f4;
typedef __attribute__((ext_vector_type(8))) float v8f;

// ---- lane utilities: keep reductions off the bpermute path ----
template <int PATTERN>
__device__ __forceinline__ float swizzle_xor(float v) {
  // group-of-32 swizzle: offset = (xor_mask<<10) | and_mask(0x1f)
  return __int_as_float(__builtin_amdgcn_ds_swizzle(__float_as_int(v), PATTERN));
}
__device__ __forceinline__ float readlane_f(float v, int lane) {
  return __int_as_float(__builtin_amdgcn_readlane(__float_as_int(v), lane));
}

// ---------------------------------------------------------------------
// Wave-wide (32 lane) sum of TWO independent f32 values with one
// V_WMMA_F32_16X16X4_F32 (all-f32, exact).
//   A (16x4): VGPR0 = p0, VGPR1 = p1 -> A[m,{0..3}] = {p0[m], p1[m], p0[m+16], p1[m+16]}
//   B (4x16): even columns select k=0,2 (p0); odd columns select k=1,3 (p1)
//   D[m, even n] = p0[m]+p0[m+16];  D[m, odd n] = p1[m]+p1[m+16]
// Sum the 8 D VGPRs, fold lane halves (SWAPX16), broadcast lanes 0/1 via
// v_readlane -> results are wave-uniform SGPR values.
// ---------------------------------------------------------------------
__device__ __forceinline__ void wave_sum_pair(float p0, float p1, int lane,
                                              float& s0, float& s1) {
  f2 a; a[0] = p0; a[1] = p1;
  float sel = (lane & 1) ? 0.0f : 1.0f;   // 1 on even columns
  f2 b; b[0] = sel; b[1] = 1.0f - sel;
  v8f c = {0.f, 0.f, 0.f, 0.f, 0.f, 0.f, 0.f, 0.f};
  v8f d = __builtin_amdgcn_wmma_f32_16x16x4_f32(
      /*neg_a=*/false, a, /*neg_b=*/false, b,
      /*c_mod=*/(short)0, c, /*reuse_a=*/false, /*reuse_b=*/false);
  float t = ((d[0] + d[1]) + (d[2] + d[3])) + ((d[4] + d[5]) + (d[6] + d[7]));
  t += swizzle_xor<0x401F>(t);            // xor 16: fold M=0..7 with M=8..15 halves
  s0 = readlane_f(t, 0);
  s1 = readlane_f(t, 1);
}

__global__ void __launch_bounds__(THREADS)
entmax_nsect_kernel(const float* __restrict__ X, float* __restrict__ out) {
  __shared__ __align__(16) float srow[DCOLS];   // 32 KB row stage
  __shared__ float sred[64];                    // cross-wave scratch

  const int tid  = threadIdx.x;
  const int lane = tid & 31;
  const int wave = tid >> 5;
  const int row  = blockIdx.x;

  const float* rowp = X + (size_t)row * DCOLS;
  float*       orow = out + (size_t)row * DCOLS;

  // ---- stage row into LDS with CDNA5 async global->LDS (ASYNCcnt) ----
  {
    uint32_t lds0 = (uint32_t)(uintptr_t)(&srow[0]);   // low 32 bits = LDS offset
#pragma unroll
    for (int i = 0; i < CHUNKS; ++i) {
      uint32_t goff  = (uint32_t)((i * THREADS + tid) * 16);  // byte offset of float4
      uint32_t laddr = lds0 + goff;
      asm volatile("global_load_async_to_lds_b128 %0, %1, %2"
                   :: "v"(laddr), "v"(goff), "s"(rowp)
                   : "memory");
    }
    asm volatile("s_wait_asynccnt 0" ::: "memory");
  }
  __syncthreads();

  // ---- LDS -> registers as native float2 pairs, pre-scaled by (alpha-1) ----
  // Keeping x packed in aligned VGPR pairs lets the hot loop run on
  // v_pk_add_f32(neg)/v_pk_fma_f32 with NO per-element packing movs.
  f2 xp[PAIRS];
#pragma unroll
  for (int i = 0; i < CHUNKS; ++i) {
    f4 v = *reinterpret_cast<const f4*>(&srow[(i * THREADS + tid) * 4]);
    f2 lo2; lo2[0] = v.x * ALPHA_M1; lo2[1] = v.y * ALPHA_M1;
    f2 hi2; hi2[0] = v.z * ALPHA_M1; hi2[1] = v.w * ALPHA_M1;
    xp[2 * i + 0] = lo2;
    xp[2 * i + 1] = hi2;
  }

  // ---- row max (thread -> wave via ds_swizzle xor -> block via LDS) ----
  float mx = fmaxf(xp[0][0], xp[0][1]);
#pragma unroll
  for (int i = 1; i < PAIRS; ++i) mx = fmaxf(mx, fmaxf(xp[i][0], xp[i][1]));
  mx = fmaxf(mx, swizzle_xor<0x041F>(mx));   // xor 1
  mx = fmaxf(mx, swizzle_xor<0x081F>(mx));   // xor 2
  mx = fmaxf(mx, swizzle_xor<0x101F>(mx));   // xor 4
  mx = fmaxf(mx, swizzle_xor<0x201F>(mx));   // xor 8
  mx = fmaxf(mx, swizzle_xor<0x401F>(mx));   // xor 16
  if (lane == 0) sred[wave] = mx;
  __syncthreads();
  float m = sred[0];
#pragma unroll
  for (int w = 1; w < 8; ++w) m = fmaxf(m, sred[w]);
  __syncthreads();

  float lo = m - 1.0f;
  float hi = m - HI_OFF;

  // ---- 5 rounds of 5-section search; 4 candidate taus per round ----
#pragma unroll
  for (int it = 0; it < N_ITER; ++it) {
    const float width = hi - lo;
    // broadcast each tau into a packed pair ONCE per round
    f2 t1; t1[0] = t1[1] = lo + width * 0.2f;
    f2 t2; t2[0] = t2[1] = lo + width * 0.4f;
    f2 t3; t3[0] = t3[1] = lo + width * 0.6f;
    f2 t4; t4[0] = t4[1] = lo + width * 0.8f;

    f2 a0 = {0.f, 0.f}, a1 = {0.f, 0.f}, a2 = {0.f, 0.f}, a3 = {0.f, 0.f};
#pragma unroll
    for (int i = 0; i < PAIRS; ++i) {
      const f2 xi = xp[i];
      f2 d0 = xi - t1; d0[0] = fmaxf(d0[0], 0.f); d0[1] = fmaxf(d0[1], 0.f);
      f2 d1 = xi - t2; d1[0] = fmaxf(d1[0], 0.f); d1[1] = fmaxf(d1[1], 0.f);
      f2 d2 = xi - t3; d2[0] = fmaxf(d2[0], 0.f); d2[1] = fmaxf(d2[1], 0.f);
      f2 d3 = xi - t4; d3[0] = fmaxf(d3[0], 0.f); d3[1] = fmaxf(d3[1], 0.f);
      a0 = d0 * d0 + a0;          // v_pk_fma_f32
      a1 = d1 * d1 + a1;
      a2 = d2 * d2 + a2;
      a3 = d3 * d3 + a3;
    }
    const float p0 = a0[0] + a0[1];
    const float p1 = a1[0] + a1[1];
    const float p2 = a2[0] + a2[1];
    const float p3 = a3[0] + a3[1];

    // wave-level sums via the matrix pipe (2 WMMAs for 4 sums)
    float s0, s1, s2, s3;
    wave_sum_pair(p0, p1, lane, s0, s1);
    wave_sum_pair(p2, p3, lane, s2, s3);

    if (lane == 0) {
      sred[wave * 4 + 0] = s0;
      sred[wave * 4 + 1] = s1;
      sred[wave * 4 + 2] = s2;
      sred[wave * 4 + 3] = s3;
    }
    __syncthreads();
    float S0 = 0.f, S1 = 0.f, S2 = 0.f, S3 = 0.f;
#pragma unroll
    for (int w = 0; w < 8; ++w) {
      S0 += sred[w * 4 + 0];
      S1 += sred[w * 4 + 1];
      S2 += sred[w * 4 + 2];
      S3 += sred[w * 4 + 3];
    }
    __syncthreads();   // sred reused next round

    // f(tau) = S - 1 decreasing in tau; count candidates with f >= 0
    float c = 0.f;
    c += (S0 >= 1.0f) ? 1.0f : 0.0f;
    c += (S1 >= 1.0f) ? 1.0f : 0.0f;
    c += (S2 >= 1.0f) ? 1.0f : 0.0f;
    c += (S3 >= 1.0f) ? 1.0f : 0.0f;

    const float new_lo = lo + width * (c * 0.2f);
    const float new_hi = lo + width * ((c + 1.0f) * 0.2f);
    lo = new_lo;
    hi = new_hi;
  }

  // ---- final tau, p = clip(Xs - tau)^2, renormalize ----
  const float tau = 0.5f * (lo + hi);
  f2 tv; tv[0] = tv[1] = tau;
  f2 psv = {0.f, 0.f};
#pragma unroll
  for (int i = 0; i < PAIRS; ++i) {
    f2 d = xp[i] - tv;
    d[0] = fmaxf(d[0], 0.f);
    d[1] = fmaxf(d[1], 0.f);
    f2 p = d * d;                 // v_pk_mul_f32
    xp[i] = p;                    // registers now hold unnormalized p
    psv = p + psv;                // v_pk_add_f32
  }
  const float ps = psv[0] + psv[1];

  float sA, sB;
  wave_sum_pair(ps, 0.0f, lane, sA, sB);
  if (lane == 0) sred[wave] = sA;
  __syncthreads();
  float S = 0.f;
#pragma unroll
  for (int w = 0; w < 8; ++w) S += sred[w];

  const float inv = 1.0f / S;

  // write-once output: non-temporal 128-bit stores (don't pollute L2)
#pragma unroll
  for (int i = 0; i < CHUNKS; ++i) {
    const f2 a = xp[2 * i + 0];
    const f2 b = xp[2 * i + 1];
    f4 v;
    v.x = a[0] * inv;
    v.y = a[1] * inv;
    v.z = b[0] * inv;
    v.w = b[1] * inv;
    __builtin_nontemporal_store(v, reinterpret_cast<f4*>(&orow[(i * THREADS + tid) * 4]));
  }
}

extern "C" void kernel_launch(void* const* d_in, const int* in_sizes, int n_in,
                              void* d_out, int out_size, void* d_ws, size_t ws_size,
                              hipStream_t stream) {
  (void)in_sizes; (void)n_in; (void)out_size; (void)d_ws; (void)ws_size;
  const float* X = (const float*)d_in[0];
  float* out = (float*)d_out;
  entmax_nsect_kernel<<<ROWS, THREADS, 0, stream>>>(X, out);
}